// EneSc_68410239091028
// MI455X (gfx1250) — compile-verified
//
#include <hip/hip_runtime.h>
#include <hip/hip_bf16.h>
#include <math.h>

// ---------------------------------------------------------------------------
// EneSc: r = ||sum_t w_t E_s[t]||^2 / ||sum_t E_s[t]||^2 ; r_th = MLP(E_q)
// Memory-bound streaming reduction over 256 MB of f32 (floor ~11us @23.3TB/s).
// Reduction + MLP run on V_WMMA_F32_16X16X4_F32 (CDNA5 matrix pipe, wave32).
// ---------------------------------------------------------------------------

typedef __attribute__((ext_vector_type(2))) float v2f;
typedef __attribute__((ext_vector_type(8))) float v8f;

#define T_ROWS 16384
#define D_COLS 4096
#define QDIM   256
#define HDIM   128

#define ROWS_PER_CHUNK 256   // rows handled per block (64 WMMA k-steps)
#define WAVES_PER_BLOCK 8    // 8 waves * 16 cols = 128 cols per block
#define THREADS 256

// ---------------------------------------------------------------- kernel 0 --
__global__ void EneSc_zero_ws(float* __restrict__ ws, int n) {
    int i = blockIdx.x * blockDim.x + threadIdx.x;
    if (i < n) ws[i] = 0.0f;
}

// ---------------------------------------------------------------- kernel 1 --
// Each wave accumulates, for a 16-column tile over ROWS_PER_CHUNK rows:
//   C[0][n] += sum_k E[t+k][n]           (A row 0 = ones)
//   C[1][n] += sum_k w[t+k] * E[t+k][n]  (A row 1 = attention weights)
__global__ void __launch_bounds__(THREADS)
EneSc_accumulate(const float* __restrict__ E,
                 const float* __restrict__ attw,
                 float* __restrict__ s_ws,
                 float* __restrict__ c_ws) {
    __shared__ float lw[ROWS_PER_CHUNK];

    const int t0 = blockIdx.y * ROWS_PER_CHUNK;
    lw[threadIdx.x] = attw[t0 + threadIdx.x];     // 256 threads, 256 weights
    __syncthreads();

    const int lane = threadIdx.x & 31;
    const int wave = threadIdx.x >> 5;
    const int m    = lane & 15;                   // A-row / B-column index
    const int half = lane >> 4;                   // K half: 0 -> K{0,1}, 1 -> K{2,3}
    const int col0 = (blockIdx.x * WAVES_PER_BLOCK + wave) * 16;

    // per-lane base: column (col0 + m) of row t0
    const float* Ebase = E + (size_t)t0 * D_COLS + (size_t)(col0 + m);

    v8f c = {};                                   // 16x16 f32 accumulator
#pragma unroll 4
    for (int i = 0; i < ROWS_PER_CHUNK / 4; ++i) {
        const int krow = 4 * i + 2 * half;        // absolute K row within chunk
        const float* p = Ebase + (size_t)krow * D_COLS;

        // B (4x16): V0 = rows K=0/K=2, V1 = rows K=1/K=3 (lane halves)
        v2f b;
        b.x = __builtin_nontemporal_load(p);           // streaming: > L2 size
        b.y = __builtin_nontemporal_load(p + D_COLS);

        // A (16x4): row 0 = 1.0, row 1 = w[t+K], rows 2..15 = 0
        const float wa = lw[krow];
        const float wb = lw[krow + 1];
        v2f a;
        a.x = (m == 0) ? 1.0f : ((m == 1) ? wa : 0.0f);
        a.y = (m == 0) ? 1.0f : ((m == 1) ? wb : 0.0f);

        c = __builtin_amdgcn_wmma_f32_16x16x4_f32(
                /*neg_a=*/false, a, /*neg_b=*/false, b,
                /*c_mod=*/(short)0, c, /*reuse_a=*/false, /*reuse_b=*/false);
    }

    // C layout: VGPR r, lanes 0-15 hold M=r (N=lane). Rows 0 and 1 are ours.
    if (half == 0) {
        atomicAdd(&s_ws[col0 + m], c[0]);   // plain column sum
        atomicAdd(&c_ws[col0 + m], c[1]);   // attention-weighted column sum
    }
}

// ---------------------------------------------------------------- kernel 2 --
// One block: energies + tiny MLP (W1 @ E_q via WMMA) + sigmoid + output.
__global__ void __launch_bounds__(THREADS)
EneSc_finalize(const float* __restrict__ s_ws,
               const float* __restrict__ c_ws,
               const float* __restrict__ Eq,
               const float* __restrict__ W1,
               const float* __restrict__ b1,
               const float* __restrict__ W2,
               const float* __restrict__ b2,
               float* __restrict__ out) {
    __shared__ float red0[THREADS];
    __shared__ float red1[THREADS];
    __shared__ float hbuf[HDIM];

    const int tid = threadIdx.x;

    // ---- energies: es = sum s^2, ec = sum c^2 over D=4096 -----------------
    float es = 0.0f, ec = 0.0f;
    for (int d = tid; d < D_COLS; d += THREADS) {
        const float sv = s_ws[d];
        const float cv = c_ws[d];
        es += sv * sv;
        ec += cv * cv;
    }
    red0[tid] = es;
    red1[tid] = ec;
    __syncthreads();
    for (int st = THREADS / 2; st > 0; st >>= 1) {
        if (tid < st) {
            red0[tid] += red0[tid + st];
            red1[tid] += red1[tid + st];
        }
        __syncthreads();
    }
    const float es_tot = red0[0];
    const float ec_tot = red1[0];
    __syncthreads();   // red0 is reused below

    // ---- h = relu(W1 @ E_q + b1) via WMMA f32 16x16x4 ---------------------
    // wave j owns W1 rows [16j, 16j+16); K advances by 4; E_q sits in B col 0.
    const int lane = tid & 31;
    const int wave = tid >> 5;
    const int m    = lane & 15;
    const int half = lane >> 4;

    v8f c = {};
    const float* Wrow = W1 + (size_t)(16 * wave + m) * QDIM;
#pragma unroll 4
    for (int k = 0; k < QDIM; k += 4) {
        const int ks = k + 2 * half;
        v2f a;                                   // A = W1 tile (16x4)
        a.x = Wrow[ks];
        a.y = Wrow[ks + 1];
        const float q0 = Eq[ks];
        const float q1 = Eq[ks + 1];
        v2f b;                                   // B col 0 = E_q, rest zero
        b.x = (m == 0) ? q0 : 0.0f;
        b.y = (m == 0) ? q1 : 0.0f;
        c = __builtin_amdgcn_wmma_f32_16x16x4_f32(
                false, a, false, b, (short)0, c, false, false);
    }
    // result column N=0: lane 0 holds M=0..7 in c[0..7], lane 16 holds M=8..15
    if (m == 0) {
        const int mbase = 16 * wave + 8 * half;
#pragma unroll
        for (int r = 0; r < 8; ++r) {
            const float hv = c[r] + b1[mbase + r];
            hbuf[mbase + r] = hv > 0.0f ? hv : 0.0f;
        }
    }
    __syncthreads();

    // ---- z = W2 @ h + b2 ; r_th = sigmoid(z) ------------------------------
    red0[tid] = (tid < HDIM) ? W2[tid] * hbuf[tid] : 0.0f;
    __syncthreads();
    for (int st = THREADS / 2; st > 0; st >>= 1) {
        if (tid < st) red0[tid] += red0[tid + st];
        __syncthreads();
    }
    if (tid == 0) {
        const float z = red0[0] + b2[0];
        out[0] = ec_tot / es_tot;               // r
        out[1] = 1.0f / (1.0f + expf(-z));      // r_th
    }
}

// ---------------------------------------------------------------- launcher --
extern "C" void kernel_launch(void* const* d_in, const int* in_sizes, int n_in,
                              void* d_out, int out_size, void* d_ws, size_t ws_size,
                              hipStream_t stream) {
    (void)in_sizes; (void)n_in; (void)out_size; (void)ws_size;

    const float* E_s  = (const float*)d_in[0];
    const float* E_q  = (const float*)d_in[1];
    const float* Attw = (const float*)d_in[2];
    const float* W1   = (const float*)d_in[3];
    const float* b1   = (const float*)d_in[4];
    const float* W2   = (const float*)d_in[5];
    const float* b2   = (const float*)d_in[6];
    float* out  = (float*)d_out;
    float* s_ws = (float*)d_ws;          // [D_COLS] plain column sums
    float* c_ws = s_ws + D_COLS;         // [D_COLS] weighted column sums

    EneSc_zero_ws<<<(2 * D_COLS + THREADS - 1) / THREADS, THREADS, 0, stream>>>(
        s_ws, 2 * D_COLS);

    dim3 grid(D_COLS / (16 * WAVES_PER_BLOCK),   // 32 column groups
              T_ROWS / ROWS_PER_CHUNK);          // 64 row chunks
    EneSc_accumulate<<<grid, THREADS, 0, stream>>>(E_s, Attw, s_ws, c_ws);

    EneSc_finalize<<<1, THREADS, 0, stream>>>(s_ws, c_ws, E_q, W1, b1, W2, b2, out);
}